// SHGNN_67164698574976
// MI455X (gfx1250) — compile-verified
//
#include <hip/hip_runtime.h>
#include <math.h>

#define N_NODES 100000
#define N_EDGES 50000
#define D_FEAT  128
#define D_FINAL 512   // 128 + 128 + 256 after 2 concat layers
#define HID     256
#define N_CLASS 40
#define NPAD    48    // N_CLASS padded to 3 WMMA tiles

#define LDA 516       // 512 + 4 pad (floats); 2064 B row stride (16B multiple)
#define LDH 260       // 256 + 4 pad

typedef __attribute__((ext_vector_type(2))) float v2f;
typedef __attribute__((ext_vector_type(8))) float v8f;

// ---------------- init: copy node_x (100K x 128) into X[:, 0:128], float4 ----------------
__global__ void shgnn_init_x(const float4* __restrict__ in, float4* __restrict__ X4) {
  int i = blockIdx.x * blockDim.x + threadIdx.x;   // over 100K*32 float4 chunks
  if (i < N_NODES * (D_FEAT / 4)) {
    int row = i >> 5;        // / 32 chunks per input row
    int c   = i & 31;
    X4[(size_t)row * (D_FINAL / 4) + c] = in[i];
  }
}

// ---------------- fixed fan-in gather + mean + relu (wave per segment, float4) ---------
// block = 256 threads = 8 waves = 8 segments; chunks = D/4 float4 per row
template <int FAN>
__global__ __launch_bounds__(256) void shgnn_gather_mean_relu(
    const float* __restrict__ src, int src_ld4,
    const int* __restrict__ map,
    float* __restrict__ dst, int dst_ld4, int dst_off4, int chunks) {
  const int lane = threadIdx.x & 31;
  const int wave = threadIdx.x >> 5;
  const int seg  = blockIdx.x * 8 + wave;

  int idx[FAN];
#pragma unroll
  for (int j = 0; j < FAN; ++j) idx[j] = map[(size_t)seg * FAN + j];

  const float4* s4 = (const float4*)src;
  float4* d4 = (float4*)dst;
  const float inv = 1.0f / (float)FAN;

  for (int c = lane; c < chunks; c += 32) {
    float4 acc = {0.f, 0.f, 0.f, 0.f};
#pragma unroll
    for (int j = 0; j < FAN; ++j) {
      float4 v = s4[(size_t)idx[j] * src_ld4 + c];
      acc.x += v.x; acc.y += v.y; acc.z += v.z; acc.w += v.w;
    }
    float4 r;
    r.x = fmaxf(acc.x * inv, 0.f);
    r.y = fmaxf(acc.y * inv, 0.f);
    r.z = fmaxf(acc.z * inv, 0.f);
    r.w = fmaxf(acc.w * inv, 0.f);
    d4[(size_t)seg * dst_ld4 + dst_off4 + c] = r;
  }
}

// ---------------- fused head: relu(X@W1+b1) @ W2 + b2 -> log_softmax ----------------
// One block (128 threads = 4 waves) per 16-row tile of X.
__global__ __launch_bounds__(128) void shgnn_head(
    const float* __restrict__ X,
    const float* __restrict__ W1, const float* __restrict__ b1,
    const float* __restrict__ W2, const float* __restrict__ b2,
    float* __restrict__ out) {
  __shared__ float As[16 * LDA];   // 16 x 512 A tile (padded)
  __shared__ float Hs[16 * LDH];   // 16 x 256 hidden tile (padded)
  __shared__ float Ls[16 * NPAD];  // 16 x 48 logits

  const int tid  = threadIdx.x;
  const int lane = tid & 31;
  const int wave = tid >> 5;
  const int m0   = blockIdx.x * 16;

  // ---- stage 16x512 A tile into LDS via gfx1250 async global->LDS copies ----
  // (lane provides LDS byte offset in a VGPR and 64-bit global address; tracked by ASYNCcnt)
  {
    const float* gbase = X + (size_t)m0 * D_FINAL;
    for (int i = tid; i < 16 * (D_FINAL / 4); i += 128) {
      int r  = i >> 7;           // 128 float4 chunks per row
      int c4 = i & 127;
      unsigned long long ga = (unsigned long long)(uintptr_t)(gbase + (size_t)r * D_FINAL + c4 * 4);
      unsigned int la = (unsigned int)(uintptr_t)(As + r * LDA + c4 * 4);  // low 32b = LDS offset
      asm volatile("global_load_async_to_lds_b128 %0, %1, off"
                   :: "v"(la), "v"(ga) : "memory");
    }
    asm volatile("s_wait_asynccnt 0x0" ::: "memory");
  }
  __syncthreads();

  const int mrow  = lane & 15;            // A-frag row
  const int khalf = (lane >> 4) * 2;      // 0 or 2 (K sub-column pair)

  // ---- GEMM1: (16x512) @ (512x256); each wave owns 4 N-tiles (64 cols) ----
  v8f acc0 = {}, acc1 = {}, acc2 = {}, acc3 = {};
  const int ncol = (lane & 15);
  for (int k = 0; k < D_FINAL; k += 4) {
    v2f a;
    const float* ap = As + mrow * LDA + k + khalf;
    a.x = ap[0]; a.y = ap[1];
    const float* bp = W1 + (size_t)(k + khalf) * HID + wave * 64 + ncol;
    v2f bA; bA.x = bp[0];  bA.y = bp[HID];
    v2f bB; bB.x = bp[16]; bB.y = bp[HID + 16];
    v2f bC; bC.x = bp[32]; bC.y = bp[HID + 32];
    v2f bD; bD.x = bp[48]; bD.y = bp[HID + 48];
    acc0 = __builtin_amdgcn_wmma_f32_16x16x4_f32(false, a, false, bA, (short)0, acc0, false, false);
    acc1 = __builtin_amdgcn_wmma_f32_16x16x4_f32(false, a, false, bB, (short)0, acc1, false, false);
    acc2 = __builtin_amdgcn_wmma_f32_16x16x4_f32(false, a, false, bC, (short)0, acc2, false, false);
    acc3 = __builtin_amdgcn_wmma_f32_16x16x4_f32(false, a, false, bD, (short)0, acc3, false, false);
  }
  // epilogue: bias + relu -> Hs
  {
    const int row_off = (lane >> 4) * 8;
    const int cb = wave * 64 + ncol;
#pragma unroll
    for (int r = 0; r < 8; ++r) {
      int row = r + row_off;
      Hs[row * LDH + cb +  0] = fmaxf(acc0[r] + b1[cb],      0.f);
      Hs[row * LDH + cb + 16] = fmaxf(acc1[r] + b1[cb + 16], 0.f);
      Hs[row * LDH + cb + 32] = fmaxf(acc2[r] + b1[cb + 32], 0.f);
      Hs[row * LDH + cb + 48] = fmaxf(acc3[r] + b1[cb + 48], 0.f);
    }
  }
  __syncthreads();

  // ---- GEMM2: (16x256) @ (256x48 padded from 40); waves 0..2 each take one N-tile ----
  if (wave < 3) {
    v8f acc = {};
    const int n = wave * 16 + ncol;
    const bool valid = (n < N_CLASS);
    for (int k = 0; k < HID; k += 4) {
      v2f a;
      const float* ap = Hs + mrow * LDH + k + khalf;
      a.x = ap[0]; a.y = ap[1];
      const int kb = k + khalf;
      v2f b;
      b.x = valid ? W2[(size_t)kb       * N_CLASS + n] : 0.f;
      b.y = valid ? W2[(size_t)(kb + 1) * N_CLASS + n] : 0.f;
      acc = __builtin_amdgcn_wmma_f32_16x16x4_f32(false, a, false, b, (short)0, acc, false, false);
    }
    const int row_off = (lane >> 4) * 8;
    const float bias = valid ? b2[n] : 0.f;
#pragma unroll
    for (int r = 0; r < 8; ++r) {
      Ls[(r + row_off) * NPAD + n] = acc[r] + bias;
    }
  }
  __syncthreads();

  // ---- log_softmax over 40 classes per row; 8 lanes per row, 5 cols each ----
  {
    const int row = tid >> 3;   // 0..15
    const int sub = tid & 7;    // 0..7
    float v[5];
    float mx = -INFINITY;
#pragma unroll
    for (int i = 0; i < 5; ++i) {
      v[i] = Ls[row * NPAD + sub + 8 * i];
      mx = fmaxf(mx, v[i]);
    }
    for (int s = 1; s < 8; s <<= 1) mx = fmaxf(mx, __shfl_xor(mx, s, 8));
    float sum = 0.f;
#pragma unroll
    for (int i = 0; i < 5; ++i) sum += __expf(v[i] - mx);
    for (int s = 1; s < 8; s <<= 1) sum += __shfl_xor(sum, s, 8);
    const float lse = mx + __logf(sum);
#pragma unroll
    for (int i = 0; i < 5; ++i) {
      out[(size_t)(m0 + row) * N_CLASS + sub + 8 * i] = v[i] - lse;
    }
  }
}

extern "C" void kernel_launch(void* const* d_in, const int* in_sizes, int n_in,
                              void* d_out, int out_size, void* d_ws, size_t ws_size,
                              hipStream_t stream) {
  (void)in_sizes; (void)n_in; (void)out_size; (void)ws_size;

  // setup_inputs order: node_x, nodes_map, edge_batch, edges_map, node_batch, W1, b1, W2, b2
  const float* node_x    = (const float*)d_in[0];
  const int*   nodes_map = (const int*)d_in[1];
  const int*   edges_map = (const int*)d_in[3];
  const float* W1 = (const float*)d_in[5];
  const float* b1 = (const float*)d_in[6];
  const float* W2 = (const float*)d_in[7];
  const float* b2 = (const float*)d_in[8];
  float* out = (float*)d_out;

  // workspace layout: X (100K x 512 f32) | E (50K x 256 f32)
  float* X = (float*)d_ws;
  float* E = X + (size_t)N_NODES * D_FINAL;

  // init X[:, 0:128] = node_x  (float4 chunks)
  {
    int total4 = N_NODES * (D_FEAT / 4);
    shgnn_init_x<<<(total4 + 255) / 256, 256, 0, stream>>>((const float4*)node_x, (float4*)X);
  }

  // layer 1 (D = 128): N2E (fan 16) then E2N (fan 8) writing X[:, 128:256]
  shgnn_gather_mean_relu<16><<<N_EDGES / 8, 256, 0, stream>>>(
      X, D_FINAL / 4, nodes_map, E, 128 / 4, 0, 128 / 4);
  shgnn_gather_mean_relu<8><<<N_NODES / 8, 256, 0, stream>>>(
      E, 128 / 4, edges_map, X, D_FINAL / 4, 128 / 4, 128 / 4);

  // layer 2 (D = 256): reads X[:, 0:256], writes X[:, 256:512]
  shgnn_gather_mean_relu<16><<<N_EDGES / 8, 256, 0, stream>>>(
      X, D_FINAL / 4, nodes_map, E, 256 / 4, 0, 256 / 4);
  shgnn_gather_mean_relu<8><<<N_NODES / 8, 256, 0, stream>>>(
      E, 256 / 4, edges_map, X, D_FINAL / 4, 256 / 4, 256 / 4);

  // fused head: one block per 16 rows
  shgnn_head<<<N_NODES / 16, 128, 0, stream>>>(X, W1, b1, W2, b2, out);
}